// KFLoss_47528108098029
// MI455X (gfx1250) — compile-verified
//
#include <hip/hip_runtime.h>
#include <stdint.h>

// ---------------- configuration ----------------
#define THREADS          256
#define TILE_BOXES       1024                       // boxes per tile per block
#define FLOATS_PER_TILE  (TILE_BOXES * 5)           // 5120 floats = 20KB
#define VEC4_PER_TILE    (FLOATS_PER_TILE / 4)      // 1280 float4 per array
#define VEC4_PER_THREAD  (VEC4_PER_TILE / THREADS)  // 5
#define BOXES_PER_THREAD (TILE_BOXES / THREADS)     // 4
#define NBLOCKS          1024

typedef float v2f __attribute__((ext_vector_type(2)));
typedef float v8f __attribute__((ext_vector_type(8)));

// ------------- fast HW math (v_rcp_f32 / v_sqrt_f32 / v_log_f32 / v_sin_f32) ---
__device__ __forceinline__ float frcp(float x)  { return __builtin_amdgcn_rcpf(x); }
__device__ __forceinline__ float fsqrt(float x) { return __builtin_amdgcn_sqrtf(x); }
__device__ __forceinline__ float flog(float x)  { return __logf(x); }

// ------------- CDNA5 async global->LDS (GV form, per-lane 16B) -------------
__device__ __forceinline__ void async_ld_b128(uint32_t lds_off, const void* g) {
    asm volatile("global_load_async_to_lds_b128 %0, %1, off"
                 :: "v"(lds_off), "v"((unsigned long long)(uintptr_t)g)
                 : "memory");
}
__device__ __forceinline__ void wait_async_tile_outstanding() {
    // one full tile (2*VEC4_PER_THREAD = 10 ops/wave) may remain in flight
#if __has_builtin(__builtin_amdgcn_s_wait_asynccnt)
    __builtin_amdgcn_s_wait_asynccnt(2 * VEC4_PER_THREAD);
#else
    asm volatile("s_wait_asynccnt 10" ::: "memory");
#endif
}
__device__ __forceinline__ void wait_async_zero() {
#if __has_builtin(__builtin_amdgcn_s_wait_asynccnt)
    __builtin_amdgcn_s_wait_asynccnt(0);
#else
    asm volatile("s_wait_asynccnt 0" ::: "memory");
#endif
}

// ------------- sigma from (x,y,w,h,r) -------------
__device__ __forceinline__ void boxSigma(const float* b, float& w, float& h,
                                         float& s11, float& s12, float& s22) {
    w = fminf(fmaxf(b[2], 1e-7f), 1e7f);
    h = fminf(fmaxf(b[3], 1e-7f), 1e7f);
    float a  = 0.25f * w * w;
    float bb = 0.25f * h * h;
    float sn = __sinf(b[4]);   // v_sin_f32 path; r in [0,1) -> ~1ulp
    float cs = __cosf(b[4]);   // v_cos_f32 path
    s11 = a * cs * cs + bb * sn * sn;
    s12 = (a - bb) * sn * cs;
    s22 = a * sn * sn + bb * cs * cs;
}

// ------------- main kernel: KFIoU + per-block loss partials -------------
__global__ __launch_bounds__(THREADS)
void kf_main(const float* __restrict__ pred, const float* __restrict__ target,
             float* __restrict__ kfiou, float* __restrict__ partials, int N) {
    __shared__ float ldsP[2][FLOATS_PER_TILE];
    __shared__ float ldsT[2][FLOATS_PER_TILE];
    __shared__ float wsum[THREADS / 32];

    const int t = threadIdx.x;
    const long long maxOff = (((long long)N * 20) - 16) & ~15LL; // last aligned 16B
    const int numTiles = (N + TILE_BOXES - 1) / TILE_BOXES;

    const uint32_t lp[2] = { (uint32_t)(uintptr_t)&ldsP[0][0],
                             (uint32_t)(uintptr_t)&ldsP[1][0] };
    const uint32_t lt[2] = { (uint32_t)(uintptr_t)&ldsT[0][0],
                             (uint32_t)(uintptr_t)&ldsT[1][0] };

    auto issue = [&](int buf, int tile) {
        const long long tb = (long long)tile * (TILE_BOXES * 20);
#pragma unroll
        for (int j = 0; j < VEC4_PER_THREAD; ++j) {
            const int v = t + j * THREADS;
            long long off = tb + (long long)v * 16;
            off = off > maxOff ? maxOff : off;   // tail clamp (aligned, in-bounds)
            async_ld_b128(lp[buf] + (uint32_t)v * 16, (const char*)pred   + off);
            async_ld_b128(lt[buf] + (uint32_t)v * 16, (const char*)target + off);
        }
    };

    float lacc = 0.0f;
    int buf = 0;
    int tile = blockIdx.x;
    if (tile < numTiles) issue(0, tile);

    for (; tile < numTiles; tile += gridDim.x) {
        const int nextTile = tile + gridDim.x;
        const bool hasNext = nextTile < numTiles;
        if (hasNext) issue(buf ^ 1, nextTile);   // prefetch into other buffer
        if (hasNext) wait_async_tile_outstanding();
        else         wait_async_zero();
        __syncthreads();                          // all waves' DMA for this tile done

#pragma unroll
        for (int j = 0; j < BOXES_PER_THREAD; ++j) {
            const int lb  = t + j * THREADS;
            const int idx = tile * TILE_BOXES + lb;
            const float* bp = &ldsP[buf][lb * 5];
            const float* bt = &ldsT[buf][lb * 5];

            float wp, hp, p11, p12, p22; boxSigma(bp, wp, hp, p11, p12, p22);
            float wt, ht, t11, t12, t22; boxSigma(bt, wt, ht, t11, t12, t22);

            // xy term: d^T Sigma_t^{-1} d
            const float d1 = bp[0] - bt[0];
            const float d2 = bp[1] - bt[1];
            const float idt = frcp(t11 * t22 - t12 * t12);
            const float quad = d1 * d1 * (t22 * idt)
                             + 2.0f * d1 * d2 * (-t12 * idt)
                             + d2 * d2 * (t11 * idt);
            const float xyl = flog(quad + 1.0f);

            // KFIoU term
            const float Vbp = wp * hp, Vbt = wt * ht;
            const float a11 = p11 + t11, a12 = p12 + t12, a22 = p22 + t22;
            const float idv = frcp(a11 * a22 - a12 * a12);
            const float v11 =  a22 * idv, v12 = -a12 * idv, v22 = a11 * idv;
            const float k11 = p11 * v11 + p12 * v12;
            const float k12 = p11 * v12 + p12 * v22;
            const float k21 = p12 * v11 + p22 * v12;
            const float k22 = p12 * v12 + p22 * v22;
            const float kp11 = k11 * p11 + k12 * p12;
            const float kp12 = k11 * p12 + k12 * p22;
            const float kp21 = k21 * p11 + k22 * p12;
            const float kp22 = k21 * p12 + k22 * p22;
            float prod = (p11 - kp11) * (p22 - kp22)
                       - (p12 - kp12) * (p12 - kp21);
            prod = prod < 0.0f ? 0.0f : prod;
            const float Vb  = 4.0f * fsqrt(prod);
            const float kfi = Vb * frcp(Vbp + Vbt - 3.0f * Vb + 1e-6f); // (4-ALPHA)=1
            const float kf  = -flog(kfi + 1e-6f);
            const float loss = fmaxf(xyl + kf, 0.0f);

            const bool ok = idx < N;
            if (ok) kfiou[idx] = kfi;
            lacc += ok ? loss : 0.0f;             // select, never 0*NaN
        }
        __syncthreads();                          // reads done before buf reuse
        buf ^= 1;
    }

    // ---- wave reduction of lacc via V_WMMA_F32_16X16X4_F32, B = ones ----
    // A layout: lane m (0..15): VGPR0=A[m][0], VGPR1=A[m][1];
    //           lane m+16:      VGPR0=A[m][2], VGPR1=A[m][3]
    // With A.vgpr0 = lacc, A.vgpr1 = 0, B = 1: D[m][n] = lacc[m] + lacc[m+16]
    v2f a; a[0] = lacc; a[1] = 0.0f;
    v2f b; b[0] = 1.0f; b[1] = 1.0f;
    v8f c = {};
    v8f d = __builtin_amdgcn_wmma_f32_16x16x4_f32(
        /*neg_a=*/false, a, /*neg_b=*/false, b,
        /*c_mod=*/(short)0, c, /*reuse_a=*/false, /*reuse_b=*/false);

    // lane 0 rows M=0..7, lane 16 rows M=8..15 -> together: sum over 32 lanes
    float s8 = d[0] + d[1] + d[2] + d[3] + d[4] + d[5] + d[6] + d[7];
    float wtot = __shfl(s8, 0, 32) + __shfl(s8, 16, 32);

    const int lane = t & 31, wid = t >> 5;
    if (lane == 0) wsum[wid] = wtot;
    __syncthreads();
    if (t == 0) {
        float bsum = 0.0f;
#pragma unroll
        for (int i = 0; i < THREADS / 32; ++i) bsum += wsum[i];
        partials[blockIdx.x] = bsum;
    }
}

// ------------- final deterministic reduction -------------
__global__ __launch_bounds__(256)
void kf_reduce(const float* __restrict__ partials, float* __restrict__ out,
               int nPart, float invN) {
    __shared__ float s[256];
    float acc = 0.0f;
    for (int i = threadIdx.x; i < nPart; i += 256) acc += partials[i];
    s[threadIdx.x] = acc;
    __syncthreads();
    for (int k = 128; k > 0; k >>= 1) {
        if (threadIdx.x < k) s[threadIdx.x] += s[threadIdx.x + k];
        __syncthreads();
    }
    if (threadIdx.x == 0) out[0] = s[0] * invN;
}

extern "C" void kernel_launch(void* const* d_in, const int* in_sizes, int n_in,
                              void* d_out, int out_size, void* d_ws, size_t ws_size,
                              hipStream_t stream) {
    const float* pred   = (const float*)d_in[0];
    const float* target = (const float*)d_in[1];
    const int N = in_sizes[0] / 5;

    float* out      = (float*)d_out;   // out[0] = mean loss, out[1..N] = KFIoU
    float* partials = (float*)d_ws;    // NBLOCKS floats of scratch

    const int numTiles = (N + TILE_BOXES - 1) / TILE_BOXES;
    int nb = NBLOCKS;
    if (nb > numTiles) nb = numTiles;
    if (nb < 1) nb = 1;

    kf_main<<<nb, THREADS, 0, stream>>>(pred, target, out + 1, partials, N);
    kf_reduce<<<1, 256, 0, stream>>>(partials, out, nb, 1.0f / (float)N);
}